// GraphAttentionLayer_31370441130105
// MI455X (gfx1250) — compile-verified
//
#include <hip/hip_runtime.h>
#include <math.h>

// GAT layer: B=8, N=2048, IN=OUT=128, leaky slope 0.2
#define B_ 8
#define N_ 2048
#define D_ 128
#define NEG_SLOPE_ 0.2f
#define EPAD 17   // LDS row pitch (floats) for the 16-row e/p buffer: kills bank conflicts

typedef __attribute__((ext_vector_type(2))) float v2f;
typedef __attribute__((ext_vector_type(4))) float v4f;   // native vector for NT b128 stores
typedef __attribute__((ext_vector_type(8))) float v8f;

// D = A(16x4,f32) * B(4x16,f32) + C(16x16,f32)  -> V_WMMA_F32_16X16X4_F32
static __device__ __forceinline__ v8f wmma_f32(v2f a, v2f b, v8f c) {
  return __builtin_amdgcn_wmma_f32_16x16x4_f32(
      /*neg_a=*/false, a, /*neg_b=*/false, b,
      /*c_mod=*/(short)0, c, /*reuse_a=*/false, /*reuse_b=*/false);
}

// ---------------------------------------------------------------------------
// Kernel 1: h = x @ W  (M = B*N = 16384, K = 128, N = 128), one wave per 16
// rows. Also computes s_i = h@a1, s_j = h@a2 per row via half-wave shuffles.
// ---------------------------------------------------------------------------
__global__ void gat_h_kernel(const float* __restrict__ x,
                             const float* __restrict__ W,
                             const float* __restrict__ avec,
                             float* __restrict__ h,
                             float* __restrict__ si,
                             float* __restrict__ sj) {
  const int row0 = blockIdx.x * 16;   // flat row into (B*N)
  const int l    = threadIdx.x;       // 0..31 (wave32)
  const int hi   = l >> 4;            // lane half: K-pair select
  const int lm   = l & 15;            // M (A) / N column (B,C,D)

  v8f zero = {};
  v8f c[8];
#pragma unroll
  for (int t = 0; t < 8; ++t) c[t] = zero;

  const float* xr = x + (size_t)(row0 + lm) * D_;
#pragma unroll 4
  for (int k0 = 0; k0 < D_; k0 += 4) {
    const int ka = k0 + hi * 2;       // this lane-half holds K = ka, ka+1
    v2f a;
    a.x = xr[ka];
    a.y = xr[ka + 1];
#pragma unroll
    for (int t = 0; t < 8; ++t) {     // 8 tiles of 16 output columns
      v2f bf;
      bf.x = W[ka * D_ + t * 16 + lm];
      bf.y = W[(ka + 1) * D_ + t * 16 + lm];
      c[t] = wmma_f32(a, bf, c[t]);
    }
  }

  const float* a1 = avec;
  const float* a2 = avec + D_;
#pragma unroll
  for (int v = 0; v < 8; ++v) {       // C VGPR v = rows v (lanes 0-15), v+8 (16-31)
    float accA = 0.f, accB = 0.f;
#pragma unroll
    for (int t = 0; t < 8; ++t) {
      const float hv = c[t][v];
      accA += hv * a1[t * 16 + lm];
      accB += hv * a2[t * 16 + lm];
    }
    // reduce across the 16 lanes of each half independently
#pragma unroll
    for (int m = 1; m < 16; m <<= 1) {
      accA += __shfl_xor(accA, m, 16);
      accB += __shfl_xor(accB, m, 16);
    }
    const int row = row0 + v + hi * 8;
    if (lm == 0) {
      si[row] = accA;
      sj[row] = accB;
    }
#pragma unroll
    for (int t = 0; t < 8; ++t)       // h stays RT (hot in L2 for kernel 2)
      h[(size_t)row * D_ + t * 16 + lm] = c[t][v];
  }
}

// ---------------------------------------------------------------------------
// Kernel 2: per 16-row block of one batch:
//   e[r][m] = leakyrelu(s_i[r] + s_j[m]) masked by adj  (kept in LDS, [m][r])
//   row softmax (two-pass) -> alpha streamed to global with NT b128 stores
//   out[16x128] = alpha_block @ h_b via WMMA f32 16x16x4 (4 waves x 2 tiles)
// ---------------------------------------------------------------------------
__global__ void gat_attn_kernel(const float* __restrict__ h,
                                const float* __restrict__ si,
                                const float* __restrict__ sj,
                                const int* __restrict__ adj,
                                float* __restrict__ out,
                                float* __restrict__ alpha) {
  extern __shared__ float smem[];
  float* eL  = smem;               // [N_][EPAD]  e then p, layout [m][r]
  float* sI  = smem + N_ * EPAD;   // 16: s_i of this row block
  float* sM  = sI + 16;            // 16: row max
  float* sS  = sM + 16;            // 16: row sum
  float* red = sS + 16;            // 128: reduction scratch

  const int tid  = threadIdx.x;    // 0..127
  const int blk  = blockIdx.x;
  const int b    = blk >> 7;       // 128 row-blocks per batch (2048/16)
  const int n0   = (blk & 127) * 16;
  const int row0 = b * N_ + n0;    // flat row into (B*N)

  if (tid < 16) sI[tid] = si[row0 + tid];
  __syncthreads();

  // ---- Pass 1: masked leaky-relu scores into LDS; adj via B128 loads -----
  {
    const float* sjbp = sj + (size_t)b * N_;
    for (int m0 = tid * 4; m0 < N_; m0 += 128 * 4) {
      const v4f sj4 = *(const v4f*)(sjbp + m0);
#pragma unroll
      for (int r = 0; r < 16; ++r) {
        const int4 a4 = *(const int4*)(adj + (size_t)(n0 + r) * N_ + m0); // b128, L2-hot
        const float sir = sI[r];
        float e0 = sir + sj4.x, e1 = sir + sj4.y, e2 = sir + sj4.z, e3 = sir + sj4.w;
        e0 = (e0 >= 0.f) ? e0 : NEG_SLOPE_ * e0;
        e1 = (e1 >= 0.f) ? e1 : NEG_SLOPE_ * e1;
        e2 = (e2 >= 0.f) ? e2 : NEG_SLOPE_ * e2;
        e3 = (e3 >= 0.f) ? e3 : NEG_SLOPE_ * e3;
        eL[(m0 + 0) * EPAD + r] = (a4.x > 0) ? e0 : -INFINITY;
        eL[(m0 + 1) * EPAD + r] = (a4.y > 0) ? e1 : -INFINITY;
        eL[(m0 + 2) * EPAD + r] = (a4.z > 0) ? e2 : -INFINITY;
        eL[(m0 + 3) * EPAD + r] = (a4.w > 0) ? e3 : -INFINITY;
      }
    }
  }
  __syncthreads();

  // ---- Pass 2: row max then row sum of exp (8 thread-groups per row) -----
  const int r = tid & 15;
  const int g = tid >> 4;          // 8 groups of 256 columns each
  {
    float mx = -INFINITY;
    for (int m = g * 256; m < (g + 1) * 256; ++m)
      mx = fmaxf(mx, eL[m * EPAD + r]);
    red[g * 16 + r] = mx;
  }
  __syncthreads();
  if (g == 0) {
    float mx = red[r];
#pragma unroll
    for (int gg = 1; gg < 8; ++gg) mx = fmaxf(mx, red[gg * 16 + r]);
    sM[r] = mx;
  }
  __syncthreads();
  {
    const float mx = sM[r];
    float s = 0.f;
    for (int m = g * 256; m < (g + 1) * 256; ++m)
      s += expf(eL[m * EPAD + r] - mx);
    red[g * 16 + r] = s;
  }
  __syncthreads();
  if (g == 0) {
    float s = red[r];
#pragma unroll
    for (int gg = 1; gg < 8; ++gg) s += red[gg * 16 + r];
    sS[r] = s;
  }
  __syncthreads();

  // ---- Pass 3: normalize; stream alpha out with NT b128 stores -----------
  float mloc[16], iloc[16];
#pragma unroll
  for (int rr = 0; rr < 16; ++rr) {
    mloc[rr] = sM[rr];
    iloc[rr] = 1.0f / sS[rr];
  }
  float* arow = alpha + (size_t)row0 * N_;
  for (int m0 = tid * 4; m0 < N_; m0 += 128 * 4) {
#pragma unroll
    for (int rr = 0; rr < 16; ++rr) {
      v4f p4;
      p4.x = expf(eL[(m0 + 0) * EPAD + rr] - mloc[rr]) * iloc[rr];
      p4.y = expf(eL[(m0 + 1) * EPAD + rr] - mloc[rr]) * iloc[rr];
      p4.z = expf(eL[(m0 + 2) * EPAD + rr] - mloc[rr]) * iloc[rr];
      p4.w = expf(eL[(m0 + 3) * EPAD + rr] - mloc[rr]) * iloc[rr];
      eL[(m0 + 0) * EPAD + rr] = p4.x;
      eL[(m0 + 1) * EPAD + rr] = p4.y;
      eL[(m0 + 2) * EPAD + rr] = p4.z;
      eL[(m0 + 3) * EPAD + rr] = p4.w;
      // written once, never re-read: NT keeps adj/h resident in the 192MB L2
      __builtin_nontemporal_store(p4, (v4f*)(arow + (size_t)rr * N_ + m0));
    }
  }
  __syncthreads();

  // ---- Pass 4: out = p(16xN) @ h_b(N x 128), WMMA f32 16x16x4 ------------
  const int w  = tid >> 5;   // wave 0..3 -> d-tiles {2w, 2w+1}
  const int l  = tid & 31;
  const int hi = l >> 4;
  const int lm = l & 15;
  const int d0 = (2 * w) * 16;
  const int d1 = d0 + 16;
  const float* hb = h + (size_t)b * N_ * D_;   // L2-resident (1 MB/batch)

  v8f zero = {};
  v8f c0 = zero, c1 = zero;
  for (int k0 = 0; k0 < N_; k0 += 4) {
    const int ka = k0 + hi * 2;                // K = ka, ka+1 for this half
    v2f a;
    a.x = eL[ka * EPAD + lm];                  // A[M=lm][K=ka]   (bank-clean)
    a.y = eL[(ka + 1) * EPAD + lm];
    v2f b0, b1;
    b0.x = hb[(size_t)ka * D_ + d0 + lm];      // B[K][N] = h[m=k][d]
    b0.y = hb[(size_t)(ka + 1) * D_ + d0 + lm];
    b1.x = hb[(size_t)ka * D_ + d1 + lm];
    b1.y = hb[(size_t)(ka + 1) * D_ + d1 + lm];
    c0 = wmma_f32(a, b0, c0);                  // A fragment reused for both tiles
    c1 = wmma_f32(a, b1, c1);
  }
#pragma unroll
  for (int v = 0; v < 8; ++v) {
    const int rr = v + hi * 8;                 // C VGPR v -> rows v / v+8
    float* orow = out + (size_t)(row0 + rr) * D_;
    __builtin_nontemporal_store(c0[v], orow + d0 + lm);
    __builtin_nontemporal_store(c1[v], orow + d1 + lm);
  }
}

// ---------------------------------------------------------------------------
extern "C" void kernel_launch(void* const* d_in, const int* in_sizes, int n_in,
                              void* d_out, int out_size, void* d_ws, size_t ws_size,
                              hipStream_t stream) {
  (void)in_sizes; (void)n_in; (void)out_size; (void)ws_size;

  const float* x    = (const float*)d_in[0];   // (8,2048,128) f32
  const int*   adj  = (const int*)  d_in[1];   // (2048,2048) i32
  const float* W    = (const float*)d_in[2];   // (128,128) f32
  const float* avec = (const float*)d_in[3];   // (256,) f32

  float* out   = (float*)d_out;                       // (8,2048,128)
  float* alpha = out + (size_t)B_ * N_ * D_;          // (8,2048,2048)

  // workspace: h (8.0 MB) + s_i + s_j  (~8.2 MB total)
  float* h_ws  = (float*)d_ws;
  float* si_ws = h_ws + (size_t)B_ * N_ * D_;
  float* sj_ws = si_ws + (size_t)B_ * N_;

  gat_h_kernel<<<(B_ * N_) / 16, 32, 0, stream>>>(x, W, avec, h_ws, si_ws, sj_ws);

  const size_t lds_bytes = (size_t)(N_ * EPAD + 16 * 3 + 128) * sizeof(float); // ~137 KB
  gat_attn_kernel<<<(B_ * N_) / 16, 128, lds_bytes, stream>>>(
      h_ws, si_ws, sj_ws, adj, out, alpha);
}